// mergeLayer_76235669504205
// MI455X (gfx1250) — compile-verified
//
#include <hip/hip_runtime.h>
#include <stdint.h>

typedef __attribute__((ext_vector_type(2))) float v2f;
typedef __attribute__((ext_vector_type(8))) float v8f;

#define Tn 128
#define Bn 512
#define Dn 500
#define LSTR 80   // input-stage LDS row stride (floats): k and k+2 rows on disjoint bank windows
#define OSTR 68   // output-stage LDS row stride (floats): rows r and r+8 on disjoint banks; 272B = 16B-aligned

// ---------------- Kernel 1: per-column greedy packing metadata ----------------
// codes[t*Bn+b] = -1 (pad token, src==1) or segment id (0..n_b-1) of token t.
// Matches reference: close the previous packet when the NEXT valid token's
// length would push curr past token_len; last valid token always closes.
__global__ void pack_meta_kernel(const int* __restrict__ src,
                                 const int* __restrict__ tok_len,
                                 const int* __restrict__ token_len_p,
                                 signed char* __restrict__ codes,
                                 int* __restrict__ nlen)
{
    int b = blockIdx.x * blockDim.x + threadIdx.x;
    if (b >= Bn) return;
    const int TL = token_len_p[0];
    int curr = 0, seg = 0;
    bool any = false;
    for (int t = 0; t < Tn; ++t) {
        int s = src[t * Bn + b];            // coalesced across b
        signed char code;
        if (s == 1) {
            code = -1;
        } else {
            int l = (s == 0) ? 4 : tok_len[s];
            if (any && (curr + l > TL)) { ++seg; curr = 0; }
            curr += l;
            any = true;
            code = (signed char)seg;        // seg stays well inside int8
        }
        codes[t * Bn + b] = code;
    }
    nlen[b] = any ? (seg + 1) : 0;
}

// ---------------- Kernel 2: stable descending rank == argsort(-nlen) ---------
__global__ void rank_kernel(const int* __restrict__ nlen,
                            int* __restrict__ order,
                            float* __restrict__ merged_out)
{
    int b = blockIdx.x * blockDim.x + threadIdx.x;
    if (b >= Bn) return;
    int len = nlen[b];
    int rank = 0;
    for (int i = 0; i < Bn; ++i) {
        int li = nlen[i];
        rank += ((li > len) || (li == len && i < b)) ? 1 : 0;
    }
    order[rank] = b;                 // order[j] = source column for output col j
    merged_out[rank] = (float)len;   // merged_lengths tail of d_out
}

// ---------------- Kernel 3: out[:, j, :] = M_b x E_b via f32 WMMA -------------
// M_b row r (<nb): 1 at tokens with seg==r. Row r (>=nb): identity e_r (keeps
// original embedding). E_b slice (128 x 64 f32) async-staged into LDS; D tiles
// restaged through LDS so all HBM traffic is b128.
__global__ __launch_bounds__(256) void pack_wmma_kernel(
    const float* __restrict__ emb,
    const signed char* __restrict__ codes,
    const int* __restrict__ order,
    const int* __restrict__ nlen,
    float* __restrict__ out)
{
    __shared__ float E[Tn * LSTR];   // 40,960 B; reused (stride OSTR) for output staging
    __shared__ int   sc[Tn];

    const int j   = blockIdx.x;      // output column
    const int dg  = blockIdx.y;      // which 64-wide d-slice (0..7; 500 padded to 512)
    const int tid = threadIdx.x;

    const int b  = order[j];
    const int nb = nlen[b];

    if (tid < Tn) sc[tid] = (int)codes[tid * Bn + b];

    // ---- async-stage the 128 x 64 f32 slice of source column b into LDS ----
    {
        const unsigned ldsbase = (unsigned)(uintptr_t)(&E[0]);
        #pragma unroll
        for (int i = 0; i < 8; ++i) {
            int chunk = tid + i * 256;        // 2048 x 16B chunks
            int t  = chunk >> 4;              // row 0..127
            int cc = chunk & 15;              // 16B chunk within row slice
            int c0 = dg * 64 + cc * 4;
            if (c0 > Dn - 4) c0 = Dn - 4;     // clamp: stay in-bounds; pad cols unused
            const float* g = emb + ((size_t)t * Bn + b) * Dn + c0;
            unsigned loff = ldsbase + (unsigned)(t * (LSTR * 4) + cc * 16);
            asm volatile("global_load_async_to_lds_b128 %0, %1, off"
                         :: "v"(loff), "v"(g) : "memory");
        }
    }
    asm volatile("s_wait_asynccnt 0x0" ::: "memory");
    __syncthreads();

    const int wave  = tid >> 5;          // r-tile 0..7 (8 waves cover M=0..127)
    const int lane  = tid & 31;
    const int m16   = lane & 15;
    const int khalf = (lane >> 4) * 2;   // A/B layout: lanes 0-15 hold K0/K1, 16-31 hold K2/K3
    const int mrow  = wave * 16 + m16;   // A-matrix row (M) held by this lane

    v8f acc[4] = {};                     // 4 adjacent 16-wide d-tiles per wave

    for (int kt = 0; kt < Tn / 4; ++kt) {
        const int k0 = kt * 4 + khalf;
        const int c0 = sc[k0];
        const int c1 = sc[k0 + 1];
        // M_b entries: segment membership OR identity row for pass-through rows
        float a0 = (c0 == mrow || (mrow >= nb && k0     == mrow)) ? 1.0f : 0.0f;
        float a1 = (c1 == mrow || (mrow >= nb && k0 + 1 == mrow)) ? 1.0f : 0.0f;
        v2f A; A.x = a0; A.y = a1;
        const float* r0 = &E[k0 * LSTR];
        const float* r1 = &E[(k0 + 1) * LSTR];
        #pragma unroll
        for (int dt = 0; dt < 4; ++dt) {
            v2f Bv; Bv.x = r0[dt * 16 + m16]; Bv.y = r1[dt * 16 + m16];
            acc[dt] = __builtin_amdgcn_wmma_f32_16x16x4_f32(
                false, A, false, Bv, (short)0, acc[dt], false, false);
        }
    }

    __syncthreads();   // all waves finished reading E; safe to overwrite

    // ---- scatter D tiles into LDS (row-major, stride OSTR) ----
    // C/D layout: VGPR i -> M = rbase + i, with the upper half-wave adding 8
    const int rbase = wave * 16 + ((lane >> 4) << 3);
    #pragma unroll
    for (int dt = 0; dt < 4; ++dt) {
        #pragma unroll
        for (int i = 0; i < 8; ++i) {
            E[(rbase + i) * OSTR + dt * 16 + m16] = acc[dt][i];
        }
    }
    __syncthreads();

    // ---- coalesced b128 stores of the 128 x (<=64) output slice ----
    #pragma unroll
    for (int i = 0; i < 8; ++i) {
        int chunk = tid + i * 256;            // 2048 x 16B chunks
        int t  = chunk >> 4;                  // output row
        int cc = chunk & 15;
        int c0 = dg * 64 + cc * 4;
        if (c0 < Dn) {                        // 500 = 4*125: chunk-aligned cut
            float4 v = *(const float4*)&E[t * OSTR + cc * 4];
            *(float4*)(out + ((size_t)t * Bn + j) * Dn + c0) = v;
        }
    }
}

extern "C" void kernel_launch(void* const* d_in, const int* in_sizes, int n_in,
                              void* d_out, int out_size, void* d_ws, size_t ws_size,
                              hipStream_t stream)
{
    (void)in_sizes; (void)n_in; (void)out_size; (void)ws_size;
    const float* emb = (const float*)d_in[0];
    const int*   src = (const int*)d_in[1];
    // d_in[2] = lengths (unused by the reference)
    const int*   tok = (const int*)d_in[3];
    const int*   tlp = (const int*)d_in[4];   // token_len scalar
    float* out = (float*)d_out;

    int* nlen          = (int*)d_ws;                  // 512 ints
    int* order         = nlen + Bn;                   // 512 ints
    signed char* codes = (signed char*)(order + Bn);  // 128*512 bytes

    pack_meta_kernel<<<dim3((Bn + 255) / 256), dim3(256), 0, stream>>>(
        src, tok, tlp, codes, nlen);
    rank_kernel<<<dim3((Bn + 255) / 256), dim3(256), 0, stream>>>(
        nlen, order, out + (size_t)Tn * Bn * Dn);
    pack_wmma_kernel<<<dim3(Bn, 8), dim3(256), 0, stream>>>(
        emb, codes, order, nlen, out);
}